// P2PNet_64089501991223
// MI455X (gfx1250) — compile-verified
//
#include <hip/hip_runtime.h>
#include <hip/hip_bf16.h>
#include <math.h>
#include <stdint.h>

// Problem constants (from reference setup_inputs)
#define BB     2
#define NN     16384
#define MM     4096
#define C1     128
#define C2     256
#define CL     (C1 + C2)      // 384 interpolated channels
#define CG     1024
#define CTOT   (3 + CL + CG)  // 1411
#define KK     3
#define NCHUNK 8
#define CHUNKN (NN / NCHUNK)  // 2048 points per wave -> 128 cols per lane (7 bits)
#define BIASF  64.0f          // makes score positive; folded into WMMA C-init
#define CNTM   127u           // low-mantissa counter mask (7 bits)

typedef __attribute__((ext_vector_type(2))) float v2f;
typedef __attribute__((ext_vector_type(8))) float v8f;

static __device__ __forceinline__ uint32_t umin32(uint32_t a, uint32_t b) {
    return __builtin_elementwise_min(a, b);   // v_min_u32
}
static __device__ __forceinline__ uint32_t umax32(uint32_t a, uint32_t b) {
    return __builtin_elementwise_max(a, b);   // v_max_u32
}

// ---------------------------------------------------------------------------
// k0: p2[b][n] = |p|^2
// ---------------------------------------------------------------------------
__global__ void k0_p2(const float* __restrict__ orig, float* __restrict__ p2) {
    int t = blockIdx.x * blockDim.x + threadIdx.x;   // 0 .. B*N-1
    if (t >= BB * NN) return;
    int b = t / NN, n = t % NN;
    const float* p = orig + (size_t)b * 3 * NN;
    float x = p[0 * NN + n], y = p[1 * NN + n], z = p[2 * NN + n];
    p2[t] = x * x + y * y + z * z;
}

// ---------------------------------------------------------------------------
// k0b: featT[b][n][c] = (c<128 ? feat1[b][c][n] : feat2[b][c-128][n])
// ---------------------------------------------------------------------------
__global__ void k0b_transpose(const float* __restrict__ f1,
                              const float* __restrict__ f2,
                              float* __restrict__ ft) {
    const int total = BB * NN * CL;
    for (int t = blockIdx.x * blockDim.x + threadIdx.x; t < total;
         t += gridDim.x * blockDim.x) {
        int c = t % CL;
        int n = (t / CL) % NN;
        int b = t / (CL * NN);
        float v = (c < C1) ? f1[((size_t)b * C1 + c) * NN + n]
                           : f2[((size_t)b * C2 + (c - C1)) * NN + n];
        ft[t] = v;
    }
}

// ---------------------------------------------------------------------------
// Exact top-3 (smallest) insertion — used only OUTSIDE the hot loop.
// ---------------------------------------------------------------------------
__device__ __forceinline__ void top3_insert(float s, int n,
                                            float& d0, float& d1, float& d2,
                                            int& i0, int& i1, int& i2) {
    const bool l0 = s < d0;
    const bool l1 = s < d1;
    const bool l2 = s < d2;
    const float t2 = l2 ? s : d2;   const int u2 = l2 ? n : i2;
    const float t1 = l1 ? s : d1;   const int u1 = l1 ? n : i1;
    const float nd2 = l1 ? d1 : t2; const int ni2 = l1 ? i1 : u2;
    const float nd1 = l0 ? d0 : t1; const int ni1 = l0 ? i0 : u1;
    const float nd0 = l0 ? s : d0;  const int ni0 = l0 ? n : i0;
    d0 = nd0; d1 = nd1; d2 = nd2;
    i0 = ni0; i1 = ni1; i2 = ni2;
}

// ---------------------------------------------------------------------------
// k1: WMMA kNN, N split into NCHUNK slices per query tile.
//   score[m][n] = p2[n] - 2*q[m].p[n] + 64   (+64 from C-init; dist^2 - q2 + 64;
//   q2 and the bias cancel in both top-k ordering and normalized weights)
//   A (16x4):  lane<16 holds (K0,K1)=(-2qx,-2qy); lane>=16 holds (K2,K3)=(-2qz, 1)
//   B (4x16):  lane<16 holds (K0,K1)=(px,py);     lane>=16 holds (K2,K3)=(pz, p2)
//   C (16x16): lane%16 = column, VGPR r = row r (lanes<16) / row 8+r (lanes>=16)
// Hot-loop top-3: positive-float-as-uint keys with 7-bit counter in the low
// mantissa; maintained with pure v_min_u32/v_max_u32 (no branches possible).
// ---------------------------------------------------------------------------
__global__ void k1_knn_wmma(const float* __restrict__ orig,
                            const float* __restrict__ query,
                            const float* __restrict__ p2g,
                            float* __restrict__ candd,
                            int* __restrict__ candi) {
    const int lane = threadIdx.x & 31;
    const int wave = threadIdx.x >> 5;
    const int wid  = blockIdx.x * (blockDim.x >> 5) + wave;
    if (wid >= BB * (MM / 16) * NCHUNK) return;
    const int chunk = wid % NCHUNK;
    const int tile  = wid / NCHUNK;               // 0 .. B*M/16-1
    const int b  = tile / (MM / 16);
    const int m0 = (tile % (MM / 16)) * 16;
    const int lm = lane & 15;
    const bool hi = lane >= 16;

    const float* q  = query + (size_t)b * 3 * MM;
    const float* p  = orig  + (size_t)b * 3 * NN;
    const float* p2 = p2g   + (size_t)b * NN;

    // A tile (constant across the N loop)
    v2f Av;
    if (!hi) { Av.x = -2.0f * q[0 * MM + m0 + lm]; Av.y = -2.0f * q[1 * MM + m0 + lm]; }
    else     { Av.x = -2.0f * q[2 * MM + m0 + lm]; Av.y = 1.0f; }

    // Hoisted per-lane B base pointers -> unconditional loads in the loop
    const float* bx = hi ? (p + 2 * NN) : (p + 0 * NN);
    const float* by = hi ? p2           : (p + 1 * NN);

    // per-lane running top-3 keys for 8 rows
    uint32_t k0[8], k1[8], k2[8];
#pragma unroll
    for (int r = 0; r < 8; ++r) k0[r] = k1[r] = k2[r] = 0x7FFFFFFFu;

    const int nbeg = chunk * CHUNKN;
    for (int it = 0; it < CHUNKN / 32; ++it) {    // 2 WMMA tiles per iteration
        const int na = nbeg + it * 32 + lm;
        const int nb = na + 16;

        v2f Bva, Bvb;
        Bva.x = bx[na]; Bva.y = by[na];
        Bvb.x = bx[nb]; Bvb.y = by[nb];

        v8f Ca = {BIASF, BIASF, BIASF, BIASF, BIASF, BIASF, BIASF, BIASF};
        v8f Cb = {BIASF, BIASF, BIASF, BIASF, BIASF, BIASF, BIASF, BIASF};
        Ca = __builtin_amdgcn_wmma_f32_16x16x4_f32(false, Av, false, Bva,
                                                   (short)0, Ca, false, false);
        Cb = __builtin_amdgcn_wmma_f32_16x16x4_f32(false, Av, false, Bvb,
                                                   (short)0, Cb, false, false);

        const uint32_t ca = (uint32_t)(2 * it);
        const uint32_t cb = (uint32_t)(2 * it + 1);
#pragma unroll
        for (int r = 0; r < 8; ++r) {
            const uint32_t ua = (__float_as_uint(Ca[r]) & ~CNTM) | ca;
            k2[r] = umin32(umax32(ua, k1[r]), k2[r]);   // uses old k1
            k1[r] = umin32(umax32(ua, k0[r]), k1[r]);   // uses old k0
            k0[r] = umin32(ua, k0[r]);
            const uint32_t ub = (__float_as_uint(Cb[r]) & ~CNTM) | cb;
            k2[r] = umin32(umax32(ub, k1[r]), k2[r]);
            k1[r] = umin32(umax32(ub, k0[r]), k1[r]);
            k0[r] = umin32(ub, k0[r]);
        }
    }

    // decode keys -> (approx dist-score, exact column)
    float d0[8], d1[8], d2[8];
    int   i0[8], i1[8], i2[8];
#pragma unroll
    for (int r = 0; r < 8; ++r) {
        d0[r] = __uint_as_float(k0[r] & ~CNTM) - BIASF;
        d1[r] = __uint_as_float(k1[r] & ~CNTM) - BIASF;
        d2[r] = __uint_as_float(k2[r] & ~CNTM) - BIASF;
        i0[r] = nbeg + (int)(k0[r] & CNTM) * 16 + lm;
        i1[r] = nbeg + (int)(k1[r] & CNTM) * 16 + lm;
        i2[r] = nbeg + (int)(k2[r] & CNTM) * 16 + lm;
    }

    // merge top-3 across the 16 lanes holding each row (xor stays within halves)
#pragma unroll
    for (int mask = 1; mask < 16; mask <<= 1) {
#pragma unroll
        for (int r = 0; r < 8; ++r) {
            float e0 = __shfl_xor(d0[r], mask, 32);
            float e1 = __shfl_xor(d1[r], mask, 32);
            float e2 = __shfl_xor(d2[r], mask, 32);
            int   j0 = __shfl_xor(i0[r], mask, 32);
            int   j1 = __shfl_xor(i1[r], mask, 32);
            int   j2 = __shfl_xor(i2[r], mask, 32);
            top3_insert(e0, j0, d0[r], d1[r], d2[r], i0[r], i1[r], i2[r]);
            top3_insert(e1, j1, d0[r], d1[r], d2[r], i0[r], i1[r], i2[r]);
            top3_insert(e2, j2, d0[r], d1[r], d2[r], i0[r], i1[r], i2[r]);
        }
    }

    // lane 0 of each half writes this chunk's top-3 candidates for its 8 rows
    if (lm == 0) {
        const int rbase = hi ? 8 : 0;
#pragma unroll
        for (int r = 0; r < 8; ++r) {
            const int row = m0 + rbase + r;
            const size_t o = (((size_t)b * MM + row) * NCHUNK + chunk) * KK;
            candd[o + 0] = d0[r];  candi[o + 0] = i0[r];
            candd[o + 1] = d1[r];  candi[o + 1] = i1[r];
            candd[o + 2] = d2[r];  candi[o + 2] = i2[r];
        }
    }
}

// ---------------------------------------------------------------------------
// k1b: merge per-chunk candidates -> final top-3 idx + normalized weights.
//   exp(-(score+q2)/2)/sum == exp(-(score-score0)/2)/sum'  (q2, bias cancel)
// ---------------------------------------------------------------------------
__global__ void k1b_merge(const float* __restrict__ candd,
                          const int* __restrict__ candi,
                          int* __restrict__ out_idx,
                          float* __restrict__ out_w) {
    const int bm = blockIdx.x * blockDim.x + threadIdx.x;  // 0 .. B*M-1
    if (bm >= BB * MM) return;

    float d0 = INFINITY, d1 = INFINITY, d2 = INFINITY;
    int   i0 = 0, i1 = 0, i2 = 0;
    const size_t base = (size_t)bm * NCHUNK * KK;
#pragma unroll
    for (int c = 0; c < NCHUNK * KK; ++c)
        top3_insert(candd[base + c], candi[base + c], d0, d1, d2, i0, i1, i2);

    const float e1 = __expf(-0.5f * (d1 - d0));
    const float e2 = __expf(-0.5f * (d2 - d0));
    const float inv = 1.0f / (1.0f + e1 + e2);
    const size_t o = (size_t)bm * KK;
    out_w[o + 0] = inv;
    out_w[o + 1] = e1 * inv;
    out_w[o + 2] = e2 * inv;
    out_idx[o + 0] = i0;
    out_idx[o + 1] = i1;
    out_idx[o + 2] = i2;
}

// ---------------------------------------------------------------------------
// k2: per (b,m) build agg[1411] in LDS, then adaptive-max-pool expand to 4096
// ---------------------------------------------------------------------------
__global__ void k2_pool(const float* __restrict__ query,
                        const float* __restrict__ gfeat,
                        const float* __restrict__ featT,
                        const int* __restrict__ knn_idx,
                        const float* __restrict__ knn_w,
                        float* __restrict__ out) {
    __shared__ float agg[CTOT + 1];
    __shared__ float sw[KK];
    __shared__ int   sid[KK];

    const int bm = blockIdx.x;            // 0 .. B*M-1
    const int b = bm / MM, m = bm % MM;
    const int t = threadIdx.x;

    if (t < KK) {
        sw[t]  = knn_w[(size_t)bm * KK + t];
        sid[t] = knn_idx[(size_t)bm * KK + t];
    }
    __syncthreads();

    // interpolated channels 0..383 (contiguous gather thanks to featT layout)
    const float* FT = featT + (size_t)b * NN * CL;
    const float w0 = sw[0], w1 = sw[1], w2 = sw[2];
    const float* r0 = FT + (size_t)sid[0] * CL;
    const float* r1 = FT + (size_t)sid[1] * CL;
    const float* r2 = FT + (size_t)sid[2] * CL;
    for (int c = t; c < CL; c += 256)
        agg[3 + c] = w0 * r0[c] + w1 * r1[c] + w2 * r2[c];

    if (t < 3) agg[t] = query[((size_t)b * 3 + t) * MM + m];

    const float* g = gfeat + (size_t)b * CG;
    for (int c = t; c < CG; c += 256)
        agg[3 + CL + c] = g[c];
    __syncthreads();

    // adaptive max pool 1411 -> 4096 (window length is 1 or 2)
    float* orow = out + (size_t)bm * MM;
#pragma unroll
    for (int i = 0; i < 16; ++i) {
        const int j = t + 256 * i;
        const int start = (j * CTOT) >> 12;                  // floor(j*1411/4096)
        const int end   = ((j + 1) * CTOT + (MM - 1)) >> 12; // ceil((j+1)*1411/4096)
        float v = agg[start];
        if (end - start == 2) v = fmaxf(v, agg[start + 1]);
        orow[j] = v;
    }
}

// ---------------------------------------------------------------------------
extern "C" void kernel_launch(void* const* d_in, const int* in_sizes, int n_in,
                              void* d_out, int out_size, void* d_ws, size_t ws_size,
                              hipStream_t stream) {
    (void)in_sizes; (void)n_in; (void)out_size; (void)ws_size;
    const float* orig  = (const float*)d_in[0];   // [B,3,N]
    const float* query = (const float*)d_in[1];   // [B,3,M]
    const float* f1    = (const float*)d_in[2];   // [B,128,N]
    const float* f2    = (const float*)d_in[3];   // [B,256,N]
    const float* gfeat = (const float*)d_in[4];   // [B,1024]
    float* out = (float*)d_out;                   // [B,M,M]

    // workspace layout
    char* ws = (char*)d_ws;
    size_t off = 0;
    float* p2    = (float*)(ws + off); off += (size_t)BB * NN * 4;
    float* candd = (float*)(ws + off); off += (size_t)BB * MM * NCHUNK * KK * 4;
    int*   candi = (int*)  (ws + off); off += (size_t)BB * MM * NCHUNK * KK * 4;
    int*   kidx  = (int*)  (ws + off); off += (size_t)BB * MM * KK * 4;
    float* kw    = (float*)(ws + off); off += (size_t)BB * MM * KK * 4;
    float* featT = (float*)(ws + off); // B*N*384 floats

    // k0: p2
    k0_p2<<<(BB * NN + 255) / 256, 256, 0, stream>>>(orig, p2);

    // k0b: feature transpose into [B][N][384]
    k0b_transpose<<<2048, 256, 0, stream>>>(f1, f2, featT);

    // k1: WMMA kNN — 8 waves/block, one (16-query tile, N-chunk) per wave
    const int nwaves = BB * (MM / 16) * NCHUNK;       // 4096 waves
    k1_knn_wmma<<<nwaves / 8, 256, 0, stream>>>(orig, query, p2, candd, candi);

    // k1b: merge candidates, compute Gaussian weights
    k1b_merge<<<(BB * MM + 255) / 256, 256, 0, stream>>>(candd, candi, kidx, kw);

    // k2: fused interpolation + concat + adaptive max pool
    k2_pool<<<BB * MM, 256, 0, stream>>>(query, gfeat, featT, kidx, kw, out);
}